// AttnKspanDecoderRNN_26731876451021
// MI455X (gfx1250) — compile-verified
//
#include <hip/hip_runtime.h>
#include <hip/hip_bf16.h>
#include <math.h>

// ---------------------------------------------------------------------------
// Sizes from the reference
// ---------------------------------------------------------------------------
#define HID     1024
#define VOCAB   32000
#define LAYERS  4
#define MAXLEN  128
#define SPAN    2

typedef __attribute__((ext_vector_type(2))) float v2f;
typedef __attribute__((ext_vector_type(4))) float v4f;
typedef __attribute__((ext_vector_type(8))) float v8f;

// ---------------------------------------------------------------------------
// Workspace layout (float offsets)
// ---------------------------------------------------------------------------
#define WS_XATTN      0                        // 3072  : [emb(2048), hidden0(1024)]
#define WS_XCOMB      (WS_XATTN + 3072)        // 3072  : [emb(2048), attn_applied(1024)]
#define WS_ATTN_PART  (WS_XCOMB + 3072)        // 4*128
#define WS_ATTN_LOG   (WS_ATTN_PART + 512)     // 128
#define WS_ATTN_W     (WS_ATTN_LOG + 128)      // 128
#define WS_COMB_PART  (WS_ATTN_W + 128)        // 4*1024
#define WS_INP        (WS_COMB_PART + 4096)    // 1024  : current GRU layer input
#define WS_GI_PART    (WS_INP + 1024)          // 2*3072
#define WS_GH_PART    (WS_GI_PART + 6144)      // 2*3072
#define WS_GI         (WS_GH_PART + 6144)      // 3072
#define WS_GH         (WS_GI + 3072)           // 3072
#define WS_LOGIT_PART (WS_GH + 3072)           // 64000 (ksplit==1)
#define WS_LOGITS     (WS_LOGIT_PART + VOCAB*SPAN) // 64000
#define WS_MS         (WS_LOGITS + VOCAB*SPAN) // 4 : {max,logsum} per span row

// d_out layout: logp[64000], hidden_new[4096], attn_weights[128]
#define OUT_HIDDEN  (VOCAB * SPAN)
#define OUT_ATTNW   (VOCAB * SPAN + LAYERS * HID)

// ---------------------------------------------------------------------------
// WMMA GEMV: part[ky*N + n] = sum_{k in chunk ky} W[n,k] * x[k]
// W row-major [N,K]. One wave -> 16 output rows. wave32, V_WMMA_F32_16X16X4_F32.
//
// The K-sum is commutative, so we remap memory-K onto WMMA logical-K to get
// fully vectorized b128 loads: per iteration each lane loads 4 contiguous
// floats of its W row at K = k + 4*(lane>>4) .. +3 (and the same slice of x,
// broadcast into all 16 A rows). WMMA #0 consumes components (0,1) as its
// logical (K0,K1)|(K2,K3) pair, WMMA #1 consumes (2,3). Each memory-k is used
// exactly once with A and B consistently paired -> exact dot product.
// D row M=0 lands in C VGPR0, lanes 0-15.
// ---------------------------------------------------------------------------
__global__ __launch_bounds__(256)
void k_gemv_wmma(const float* __restrict__ W, const float* __restrict__ x,
                 float* __restrict__ part, int N, int K, int kchunk) {
  const int lane = threadIdx.x & 31;
  const int tile = blockIdx.x * (blockDim.x >> 5) + (threadIdx.x >> 5);
  const int n0   = tile << 4;
  const int k0   = blockIdx.y * kchunk;
  const int h4   = (lane >> 4) << 2;                 // 0 for lanes 0-15, 4 for 16-31
  const float* __restrict__ wrow = W + (size_t)(n0 + (lane & 15)) * (size_t)K;

  v8f c = {};
  #pragma unroll 2
  for (int k = k0; k < k0 + kchunk; k += 8) {        // kchunk % 8 == 0
    v4f xa = *(const v4f*)(x    + k + h4);           // global_load_b128
    v4f wb = *(const v4f*)(wrow + k + h4);           // global_load_b128
    v2f a0, b0, a1, b1;
    a0.x = xa.x; a0.y = xa.y; b0.x = wb.x; b0.y = wb.y;
    a1.x = xa.z; a1.y = xa.w; b1.x = wb.z; b1.y = wb.w;
    c = __builtin_amdgcn_wmma_f32_16x16x4_f32(false, a0, false, b0, (short)0, c, false, false);
    c = __builtin_amdgcn_wmma_f32_16x16x4_f32(false, a1, false, b1, (short)0, c, false, false);
  }
  if (lane < 16)
    part[(size_t)blockIdx.y * N + n0 + lane] = c[0];
}

// out[n] = bias[n] + sum_j part[j*N+n]  (optionally ReLU)
__global__ void k_reduce(const float* __restrict__ part, const float* __restrict__ bias,
                         float* __restrict__ out, int N, int ksplit, int do_relu) {
  int n = blockIdx.x * blockDim.x + threadIdx.x;
  if (n >= N) return;
  float s = bias[n];
  for (int j = 0; j < ksplit; ++j) s += part[(size_t)j * N + n];
  if (do_relu) s = fmaxf(s, 0.0f);
  out[n] = s;
}

// Gather embedding rows; build attn input [emb, h0] and comb-input prefix [emb, ...]
__global__ void k_embed(const int* __restrict__ ids, const float* __restrict__ emb,
                        const float* __restrict__ hidden,
                        float* __restrict__ xattn, float* __restrict__ xcomb) {
  int i = blockIdx.x * blockDim.x + threadIdx.x;     // 0..3071
  float v;
  if (i < SPAN * HID) {
    int tok = ids[i >> 10];
    v = emb[(size_t)tok * HID + (i & (HID - 1))];
    xcomb[i] = v;
  } else {
    v = hidden[i - SPAN * HID];                      // hidden[0]
  }
  xattn[i] = v;
}

// Softmax over 128 attention logits; write to ws and to d_out attn_weights slot
__global__ void k_softmax128(const float* __restrict__ logits,
                             float* __restrict__ w_ws, float* __restrict__ w_out) {
  __shared__ float red[MAXLEN];
  int i = threadIdx.x;
  float v = logits[i];
  red[i] = v; __syncthreads();
  for (int s = MAXLEN / 2; s > 0; s >>= 1) { if (i < s) red[i] = fmaxf(red[i], red[i + s]); __syncthreads(); }
  float m = red[0]; __syncthreads();
  float e = expf(v - m);
  red[i] = e; __syncthreads();
  for (int s = MAXLEN / 2; s > 0; s >>= 1) { if (i < s) red[i] += red[i + s]; __syncthreads(); }
  float w = e / red[0];
  w_ws[i] = w;
  w_out[i] = w;
}

// attn_applied[n] = sum_k w[k] * enc[k][n]  (coalesced over n) -> xcomb[2048+n]
__global__ void k_attn_apply(const float* __restrict__ w, const float* __restrict__ enc,
                             float* __restrict__ xcomb) {
  int n = blockIdx.x * blockDim.x + threadIdx.x;     // 0..1023
  float s = 0.0f;
  #pragma unroll 8
  for (int k = 0; k < MAXLEN; ++k) s += w[k] * enc[(size_t)k * HID + n];
  xcomb[SPAN * HID + n] = s;
}

// GRU gates (PyTorch order r,z,n); writes new h to d_out and to next-layer input
__global__ void k_gru_gates(const float* __restrict__ gi, const float* __restrict__ gh,
                            const float* __restrict__ h,
                            float* __restrict__ hnew_out, float* __restrict__ inp_next) {
  int j = blockIdx.x * blockDim.x + threadIdx.x;     // 0..1023
  float r = 1.0f / (1.0f + expf(-(gi[j]           + gh[j])));
  float z = 1.0f / (1.0f + expf(-(gi[j + HID]     + gh[j + HID])));
  float n = tanhf(gi[j + 2 * HID] + r * gh[j + 2 * HID]);
  float h2 = (1.0f - z) * n + z * h[j];
  hnew_out[j] = h2;
  inp_next[j] = h2;
}

// Per span row: max and log(sum(exp)) over VOCAB logits
__global__ void k_lse(const float* __restrict__ logits, float* __restrict__ ms) {
  __shared__ float red[256];
  const float* p = logits + (size_t)blockIdx.x * VOCAB;
  int t = threadIdx.x;
  float m = -INFINITY;
  for (int i = t; i < VOCAB; i += 256) m = fmaxf(m, p[i]);
  red[t] = m; __syncthreads();
  for (int s = 128; s > 0; s >>= 1) { if (t < s) red[t] = fmaxf(red[t], red[t + s]); __syncthreads(); }
  m = red[0]; __syncthreads();
  float acc = 0.0f;
  for (int i = t; i < VOCAB; i += 256) acc += expf(p[i] - m);
  red[t] = acc; __syncthreads();
  for (int s = 128; s > 0; s >>= 1) { if (t < s) red[t] += red[t + s]; __syncthreads(); }
  if (t == 0) { ms[blockIdx.x * 2] = m; ms[blockIdx.x * 2 + 1] = logf(red[0]); }
}

__global__ void k_write_logp(const float* __restrict__ logits, const float* __restrict__ ms,
                             float* __restrict__ out) {
  int i = blockIdx.x * blockDim.x + threadIdx.x;     // 0..63999 exact
  int row = (i < VOCAB) ? 0 : 1;
  out[i] = logits[i] - ms[row * 2] - ms[row * 2 + 1];
}

// ---------------------------------------------------------------------------
extern "C" void kernel_launch(void* const* d_in, const int* in_sizes, int n_in,
                              void* d_out, int out_size, void* d_ws, size_t ws_size,
                              hipStream_t stream) {
  const int*   ids    = (const int*)  d_in[0];
  const float* hidden = (const float*)d_in[1];   // [LAYERS, 1, HID]
  const float* enc    = (const float*)d_in[2];   // [MAXLEN, HID]
  const float* emb    = (const float*)d_in[3];   // [VOCAB, HID]
  const float* attn_W = (const float*)d_in[4];   // [MAXLEN, 3*HID]
  const float* attn_b = (const float*)d_in[5];
  const float* comb_W = (const float*)d_in[6];   // [HID, 3*HID]
  const float* comb_b = (const float*)d_in[7];
  const float* gWih   = (const float*)d_in[8];   // [LAYERS, 3*HID, HID]
  const float* gWhh   = (const float*)d_in[9];
  const float* gbih   = (const float*)d_in[10];  // [LAYERS, 3*HID]
  const float* gbhh   = (const float*)d_in[11];
  const float* out_W  = (const float*)d_in[12];  // [VOCAB*SPAN, HID]
  const float* out_b  = (const float*)d_in[13];

  float* ws  = (float*)d_ws;
  float* out = (float*)d_out;

  const int KCAT = (1 + SPAN) * HID;             // 3072

  // 1) embed gather + build attn/comb input vectors
  k_embed<<<KCAT / 256, 256, 0, stream>>>(ids, emb, hidden, ws + WS_XATTN, ws + WS_XCOMB);

  // 2) attention logits: [128 x 3072] GEMV, split-K=4
  k_gemv_wmma<<<dim3(1, 4), 256, 0, stream>>>(attn_W, ws + WS_XATTN, ws + WS_ATTN_PART,
                                              MAXLEN, KCAT, KCAT / 4);
  k_reduce<<<1, MAXLEN, 0, stream>>>(ws + WS_ATTN_PART, attn_b, ws + WS_ATTN_LOG,
                                     MAXLEN, 4, 0);
  k_softmax128<<<1, MAXLEN, 0, stream>>>(ws + WS_ATTN_LOG, ws + WS_ATTN_W, out + OUT_ATTNW);

  // 3) attn_applied = w @ enc  -> xcomb[2048:3072]
  k_attn_apply<<<HID / 256, 256, 0, stream>>>(ws + WS_ATTN_W, enc, ws + WS_XCOMB);

  // 4) combine + relu: [1024 x 3072] GEMV, split-K=4
  k_gemv_wmma<<<dim3(8, 4), 256, 0, stream>>>(comb_W, ws + WS_XCOMB, ws + WS_COMB_PART,
                                              HID, KCAT, KCAT / 4);
  k_reduce<<<HID / 256, 256, 0, stream>>>(ws + WS_COMB_PART, comb_b, ws + WS_INP,
                                          HID, 4, 1);

  // 5) GRU layers: gi = Wih @ inp, gh = Whh @ h  (each [3072 x 1024], split-K=2)
  for (int l = 0; l < LAYERS; ++l) {
    const float* Wih = gWih + (size_t)l * 3 * HID * HID;
    const float* Whh = gWhh + (size_t)l * 3 * HID * HID;
    k_gemv_wmma<<<dim3(24, 2), 256, 0, stream>>>(Wih, ws + WS_INP, ws + WS_GI_PART,
                                                 3 * HID, HID, HID / 2);
    k_gemv_wmma<<<dim3(24, 2), 256, 0, stream>>>(Whh, hidden + (size_t)l * HID, ws + WS_GH_PART,
                                                 3 * HID, HID, HID / 2);
    k_reduce<<<3 * HID / 256, 256, 0, stream>>>(ws + WS_GI_PART, gbih + (size_t)l * 3 * HID,
                                                ws + WS_GI, 3 * HID, 2, 0);
    k_reduce<<<3 * HID / 256, 256, 0, stream>>>(ws + WS_GH_PART, gbhh + (size_t)l * 3 * HID,
                                                ws + WS_GH, 3 * HID, 2, 0);
    k_gru_gates<<<HID / 256, 256, 0, stream>>>(ws + WS_GI, ws + WS_GH,
                                               hidden + (size_t)l * HID,
                                               out + OUT_HIDDEN + (size_t)l * HID,
                                               ws + WS_INP);
  }

  // 6) output projection: [64000 x 1024] GEMV (4000 waves, split-K=1)
  k_gemv_wmma<<<dim3(500, 1), 256, 0, stream>>>(out_W, ws + WS_INP, ws + WS_LOGIT_PART,
                                                VOCAB * SPAN, HID, HID);
  k_reduce<<<VOCAB * SPAN / 256, 256, 0, stream>>>(ws + WS_LOGIT_PART, out_b, ws + WS_LOGITS,
                                                   VOCAB * SPAN, 1, 0);

  // 7) log_softmax per span row
  k_lse<<<SPAN, 256, 0, stream>>>(ws + WS_LOGITS, ws + WS_MS);
  k_write_logp<<<VOCAB * SPAN / 256, 256, 0, stream>>>(ws + WS_LOGITS, ws + WS_MS, out);
}